// NodeAttAggregator_42442866819641
// MI455X (gfx1250) — compile-verified
//
#include <hip/hip_runtime.h>
#include <hip/hip_bf16.h>
#include <math.h>

// Problem dims (fixed by the reference).
#define H_EDGES 4096
#define N_NODES 20000
#define FIN     256
#define FOUT    128
#define DEG     32

typedef __attribute__((ext_vector_type(2))) float v2f;
typedef __attribute__((ext_vector_type(4))) float v4f;
typedef __attribute__((ext_vector_type(8))) float v8f;

// -----------------------------------------------------------------------------
// Kernel 1: fused projection
//   kproj[n,:] = node_embed[n,:]  @ Wq      (n = 0..19999)
//   qproj[h,:] = hedge_embed[h,:] @ Wq      (h = 0..4095)
// One 16x16 FP32 output tile per wave via V_WMMA_F32_16X16X4_F32.
// Row-tile space is the concatenation [node_embed ; hedge_embed] = 24096 rows
// = 1506 tiles of 16 (both row counts are multiples of 16 -> no tails).
// Block = 256 threads = 8 waves; wave w owns FOUT col-tile w (8 * 16 = 128).
// -----------------------------------------------------------------------------
__global__ __launch_bounds__(256) void proj_wmma_f32(
    const float* __restrict__ hedge, const float* __restrict__ node,
    const float* __restrict__ Wq, float* __restrict__ kproj,
    float* __restrict__ qproj)
{
  const int lane = threadIdx.x & 31;
  const int wave = threadIdx.x >> 5;   // 0..7 -> FOUT column tile
  const int t    = blockIdx.x;         // 0..1505 row tile in concat space

  const float* src; float* dst; int rowbase;
  if (t < N_NODES / 16) { src = node;  dst = kproj; rowbase = t * 16; }
  else                  { src = hedge; dst = qproj; rowbase = (t - N_NODES / 16) * 16; }

  const int C    = wave * 16;          // FOUT column base
  const int ar   = lane & 15;          // A row within tile / B column within tile
  const int koff = (lane >> 4) << 1;   // 0 for lanes 0-15, 2 for lanes 16-31

  // A (16x4 f32): lane L: VGPR0 = A[L%16][koff], VGPR1 = A[L%16][koff+1]
  const float* arow = src + (size_t)(rowbase + ar) * FIN + koff;
  // B (4x16 f32): lane L: VGPR0 = B[koff][C + L%16], VGPR1 = B[koff+1][C + L%16]
  const float* bcol = Wq + (size_t)koff * FOUT + C + ar;

  v8f acc = {};
  #pragma unroll 4
  for (int k0 = 0; k0 < FIN; k0 += 4) {
    v2f a = *(const v2f*)(arow + k0);              // 8B-aligned float2
    v2f b;
    b.x = bcol[(size_t)(k0    ) * FOUT];
    b.y = bcol[(size_t)(k0 + 1) * FOUT];
    // 8 args: (neg_a, A, neg_b, B, c_mod, C, reuse_a, reuse_b)
    acc = __builtin_amdgcn_wmma_f32_16x16x4_f32(
        /*neg_a=*/false, a, /*neg_b=*/false, b,
        /*c_mod=*/(short)0, acc, /*reuse_a=*/false, /*reuse_b=*/false);
  }

  // D 16x16 f32 layout: VGPR g -> M = g + (lane/16)*8, N = lane%16
  const int drow = rowbase + ((lane >> 4) << 3);
  const int dcol = C + ar;
  #pragma unroll
  for (int g = 0; g < 8; ++g)
    dst[(size_t)(drow + g) * FOUT + dcol] = acc[g];
}

// -----------------------------------------------------------------------------
// Kernel 2: per-hyperedge sparse attention + aggregation.
// One wave per hyperedge (wave32: lane = member slot j).
//   s_j   = dot(q_h, k[c_j]) / sqrt(FOUT)         (dup members -> -inf, matching
//                                                  the reference's boolean mask)
//   attn  = softmax over the 32 lanes
//   out[f,h] = sum_j attn_j * k[c_j][f]           (lanes re-striped over features
//                                                  for coalesced 128B row reads)
// Block = 256 threads = 8 hyperedges.
// -----------------------------------------------------------------------------
__global__ __launch_bounds__(256) void attn_agg(
    const float* __restrict__ qproj, const float* __restrict__ kproj,
    const int* __restrict__ col_idx, float* __restrict__ out)
{
  const int lane = threadIdx.x & 31;
  const int wave = threadIdx.x >> 5;
  const int h    = blockIdx.x * 8 + wave;

  const int c = col_idx[(size_t)h * DEG + lane];

  // score: dot(q_h, k_c) over 128 features
  const float* kr = kproj + (size_t)c * FOUT;
  const float* qr = qproj + (size_t)h * FOUT;
  float s = 0.f;
  #pragma unroll
  for (int f = 0; f < FOUT; f += 4) {
    v4f kv = *(const v4f*)(kr + f);
    v4f qv = *(const v4f*)(qr + f);
    s += kv.x * qv.x + kv.y * qv.y + kv.z * qv.z + kv.w * qv.w;
  }
  s *= 0.0883883476483184f;  // 1/sqrt(128)

  // Dedup: the reference's mask.set(True) counts duplicate members once.
  // A slot is dead if an earlier slot in the same edge names the same node.
  bool dup = false;
  #pragma unroll
  for (int i = 0; i < DEG; ++i) {
    int ci = __shfl(c, i, 32);
    if (i < lane && ci == c) dup = true;
  }
  if (dup) s = -INFINITY;

  // 32-lane softmax (lane 0 is never dup, so the max is always finite)
  float m = s;
  #pragma unroll
  for (int off = 16; off > 0; off >>= 1) m = fmaxf(m, __shfl_xor(m, off, 32));
  float e = __expf(s - m);               // dup lanes: exp(-inf) = 0
  float sum = e;
  #pragma unroll
  for (int off = 16; off > 0; off >>= 1) sum += __shfl_xor(sum, off, 32);
  const float w = e / sum;

  // Aggregation: lanes now stripe features (f = lane + 32t). Each iteration
  // broadcasts (w_i, c_i) and all 32 lanes read 128 contiguous bytes of the
  // same k row -> fully coalesced, L2-hot.
  float a0 = 0.f, a1 = 0.f, a2 = 0.f, a3 = 0.f;
  for (int i = 0; i < DEG; ++i) {
    float wi = __shfl(w, i, 32);
    int   ci = __shfl(c, i, 32);
    const float* kri = kproj + (size_t)ci * FOUT;
    a0 += wi * kri[lane];
    a1 += wi * kri[lane + 32];
    a2 += wi * kri[lane + 64];
    a3 += wi * kri[lane + 96];
  }

  // out is [FOUT, H] row-major: out[f*H + h]
  out[(size_t)(lane      ) * H_EDGES + h] = a0;
  out[(size_t)(lane + 32 ) * H_EDGES + h] = a1;
  out[(size_t)(lane + 64 ) * H_EDGES + h] = a2;
  out[(size_t)(lane + 96 ) * H_EDGES + h] = a3;
}

// -----------------------------------------------------------------------------
// Launch. Inputs (setup_inputs order):
//   d_in[0] hedge_embed [4096,256] f32
//   d_in[1] node_embed  [20000,256] f32
//   d_in[2] Wq          [256,128]  f32
//   d_in[3] row_idx     (implied layout: edge h owns col_idx[h*32 .. h*32+31])
//   d_in[4] col_idx     [131072]   int
// Workspace: kproj 20000*128 f32 (10.24MB) + qproj 4096*128 f32 (2MB).
// -----------------------------------------------------------------------------
extern "C" void kernel_launch(void* const* d_in, const int* in_sizes, int n_in,
                              void* d_out, int out_size, void* d_ws, size_t ws_size,
                              hipStream_t stream) {
  (void)in_sizes; (void)n_in; (void)out_size; (void)ws_size;
  const float* hedge   = (const float*)d_in[0];
  const float* node    = (const float*)d_in[1];
  const float* Wq      = (const float*)d_in[2];
  const int*   col_idx = (const int*)d_in[4];

  float* kproj = (float*)d_ws;
  float* qproj = kproj + (size_t)N_NODES * FOUT;

  // 1506 row tiles of 16 over [node ; hedge]; 8 waves/block cover FOUT=128.
  proj_wmma_f32<<<(N_NODES + H_EDGES) / 16, 256, 0, stream>>>(
      hedge, node, Wq, kproj, qproj);

  // 4096 hyperedges, 8 per block (one wave each).
  attn_agg<<<H_EDGES / 8, 256, 0, stream>>>(
      qproj, kproj, col_idx, (float*)d_out);
}